// StateGenerator_69861938037398
// MI455X (gfx1250) — compile-verified
//
#include <hip/hip_runtime.h>
#include <hip/hip_bf16.h>

#define B_  16
#define S_  512
#define D_  400
#define V_  32000
#define T_  8
#define NS_ 30
#define NG_ 3

typedef __attribute__((ext_vector_type(2))) float v2f;
typedef __attribute__((ext_vector_type(4))) float v4f;
typedef __attribute__((ext_vector_type(8))) float v8f;

// ---------------------------------------------------------------------------
// Non-temporal zero-fill of out[base .. base+n) in 128-bit units.
// Slices 1..NS-1 of ptr_outputs plus the gate region are permanent zeros that
// are never re-read: NT stores keep the 192MB L2 free for shared_emb/logits.
// ---------------------------------------------------------------------------
__global__ void k_zero(v4f* __restrict__ out, unsigned long long n4) {
    unsigned long long i = (unsigned long long)blockIdx.x * blockDim.x + threadIdx.x;
    unsigned long long stride = (unsigned long long)gridDim.x * blockDim.x;
    v4f z = {0.f, 0.f, 0.f, 0.f};
    for (; i < n4; i += stride) __builtin_nontemporal_store(z, &out[i]);
}

// ---------------------------------------------------------------------------
// X[b,0,:] = slot_emb; X[b,t,:] = shared_emb[trg[b,0,t-1],:]
// ---------------------------------------------------------------------------
__global__ void k_build_x(const float* __restrict__ shared_emb,
                          const float* __restrict__ slot_emb,
                          const int*   __restrict__ trg,
                          float* __restrict__ X) {
    int i = blockIdx.x * blockDim.x + threadIdx.x;        // over B*T*D
    if (i >= B_ * T_ * D_) return;
    int d = i % D_;
    int bt = i / D_;
    int t = bt % T_;
    int b = bt / T_;
    float v;
    if (t == 0) {
        v = slot_emb[d];
    } else {
        int tok = trg[(b * NS_ + 0) * T_ + (t - 1)];
        v = shared_emb[(size_t)tok * D_ + d];
    }
    X[i] = v;
}

// ---------------------------------------------------------------------------
// WMMA f32 GEMM: C[M x N] = A[M x K] * Bw[N x K]^T (+ bias[n])
// Requirements: M,N multiples of 16; K multiple of 4.
// Block = (32,4): 4 waves, each wave owns one 16x16 D tile.
// Fragment layouts per CDNA5 ISA 7.12.2 (32-bit A 16x4, 32-bit C/D 16x16).
// ---------------------------------------------------------------------------
__global__ __launch_bounds__(128)
void k_wmma_gemm(const float* __restrict__ A,
                 const float* __restrict__ Bw,
                 const float* __restrict__ bias,
                 float* __restrict__ C,
                 int M, int N, int K) {
    int ntile = blockIdx.x * blockDim.y + threadIdx.y;    // wave-uniform
    int mtile = blockIdx.y;
    if (ntile * 16 >= N) return;                          // uniform per wave -> EXEC stays full

    int lane = threadIdx.x;                               // 0..31
    int half = lane >> 4;                                 // 0: K pair {0,1}; 1: K pair {2,3}
    int l16  = lane & 15;

    const float* arow = A  + (size_t)(mtile * 16 + l16) * K + half * 2;
    const float* brow = Bw + (size_t)(ntile * 16 + l16) * K + half * 2;

    v8f acc = {};
    for (int k0 = 0; k0 < K; k0 += 4) {
        v2f a, b;
        a.x = arow[k0];     a.y = arow[k0 + 1];           // A[m, k0+half*2 (+1)]
        b.x = brow[k0];     b.y = brow[k0 + 1];           // B[k0+half*2 (+1), n] = Bw[n, ...]
        acc = __builtin_amdgcn_wmma_f32_16x16x4_f32(
                  false, a, false, b, (short)0, acc, false, false);
    }

    int   nc = ntile * 16 + l16;                          // this lane's column
    float bz = bias ? bias[nc] : 0.0f;
    int   mbase = mtile * 16 + half * 8;                  // rows half*8 .. half*8+7
#pragma unroll
    for (int r = 0; r < 8; ++r) {
        C[(size_t)(mbase + r) * N + nc] = acc[r] + bz;
    }
}

// ---------------------------------------------------------------------------
// GRU elementwise: h = (1-z)*tanh(i_n + r*h_n) + z*h0
// ---------------------------------------------------------------------------
__global__ void k_gru(const float* __restrict__ gi,   // (B*T, 3D)
                      const float* __restrict__ gh,   // (B, 3D)
                      const float* __restrict__ h0,   // (B, D)
                      float* __restrict__ h) {        // (B*T, D)
    int i = blockIdx.x * blockDim.x + threadIdx.x;
    if (i >= B_ * T_ * D_) return;
    int d  = i % D_;
    int bt = i / D_;
    int b  = bt / T_;
    const float* gib = gi + (size_t)bt * 3 * D_;
    const float* ghb = gh + (size_t)b  * 3 * D_;
    float r = 1.0f / (1.0f + __expf(-(gib[d]          + ghb[d])));
    float z = 1.0f / (1.0f + __expf(-(gib[D_ + d]     + ghb[D_ + d])));
    float n = tanhf(gib[2 * D_ + d] + r * ghb[2 * D_ + d]);
    h[i] = (1.0f - z) * n + z * h0[(size_t)b * D_ + d];
}

// ---------------------------------------------------------------------------
// scores[b,t,s] = dot(h[b,t,:], enc[b,s,:])
// ---------------------------------------------------------------------------
__global__ void k_scores(const float* __restrict__ h,
                         const float* __restrict__ enc,
                         float* __restrict__ scores) {
    int i = blockIdx.x * blockDim.x + threadIdx.x;        // over B*T*S
    if (i >= B_ * T_ * S_) return;
    int s  = i % S_;
    int bt = i / S_;
    int b  = bt / T_;
    const float* hr = h   + (size_t)bt * D_;
    const float* er = enc + ((size_t)b * S_ + s) * D_;
    float acc = 0.0f;
    for (int d = 0; d < D_; ++d) acc += hr[d] * er[d];
    scores[i] = acc;
}

// ---------------------------------------------------------------------------
// probs = softmax(scores) over S (one block per (b,t) row)
// ---------------------------------------------------------------------------
__global__ __launch_bounds__(256)
void k_softmax_s(const float* __restrict__ scores, float* __restrict__ probs) {
    int bt = blockIdx.x;
    const float* row = scores + (size_t)bt * S_;
    __shared__ float red[256];
    int tid = threadIdx.x;

    float m = -1e30f;
    for (int s = tid; s < S_; s += 256) m = fmaxf(m, row[s]);
    red[tid] = m; __syncthreads();
    for (int o = 128; o > 0; o >>= 1) {
        if (tid < o) red[tid] = fmaxf(red[tid], red[tid + o]);
        __syncthreads();
    }
    m = red[0]; __syncthreads();

    float sum = 0.0f;
    for (int s = tid; s < S_; s += 256) sum += __expf(row[s] - m);
    red[tid] = sum; __syncthreads();
    for (int o = 128; o > 0; o >>= 1) {
        if (tid < o) red[tid] += red[tid + o];
        __syncthreads();
    }
    float inv = 1.0f / red[0];
    for (int s = tid; s < S_; s += 256)
        probs[(size_t)bt * S_ + s] = __expf(row[s] - m) * inv;
}

// ---------------------------------------------------------------------------
// context[b,t,d] = sum_s scores[b,t,s] * enc[b,s,d]   (RAW scores, per reference)
// ---------------------------------------------------------------------------
__global__ void k_context(const float* __restrict__ scores,
                          const float* __restrict__ enc,
                          float* __restrict__ ctx) {
    int i = blockIdx.x * blockDim.x + threadIdx.x;        // over B*T*D
    if (i >= B_ * T_ * D_) return;
    int d  = i % D_;
    int bt = i / D_;
    int b  = bt / T_;
    const float* sr = scores + (size_t)bt * S_;
    const float* eb = enc + (size_t)b * S_ * D_ + d;
    float acc = 0.0f;
    for (int s = 0; s < S_; ++s) acc += sr[s] * eb[(size_t)s * D_];
    ctx[i] = acc;
}

// ---------------------------------------------------------------------------
// gate0[b,g] = ctx[b,0,:] . w_gate_w[g,:] + w_gate_b[g]  -> out[gate_off + b*NG + g]
// ---------------------------------------------------------------------------
__global__ void k_gate(const float* __restrict__ ctx,
                       const float* __restrict__ wg,
                       const float* __restrict__ bg,
                       float* __restrict__ out, unsigned long long gate_off) {
    int i = threadIdx.x;
    if (i >= B_ * NG_) return;
    int g = i % NG_;
    int b = i / NG_;
    const float* cr = ctx + (size_t)(b * T_ + 0) * D_;
    const float* wr = wg + (size_t)g * D_;
    float acc = bg[g];
    for (int d = 0; d < D_; ++d) acc += cr[d] * wr[d];
    out[gate_off + i] = acc;
}

// ---------------------------------------------------------------------------
// p_gen[b,t] = sigmoid( [h, ctx, X] . w_ratio_w + w_ratio_b )
// ---------------------------------------------------------------------------
__global__ void k_pgen(const float* __restrict__ h,
                       const float* __restrict__ ctx,
                       const float* __restrict__ X,
                       const float* __restrict__ wr,
                       const float* __restrict__ br,
                       float* __restrict__ pgen) {
    int bt = blockIdx.x * blockDim.x + threadIdx.x;
    if (bt >= B_ * T_) return;
    const float* hr = h   + (size_t)bt * D_;
    const float* cr = ctx + (size_t)bt * D_;
    const float* xr = X   + (size_t)bt * D_;
    float acc = br[0];
    for (int d = 0; d < D_; ++d)
        acc += hr[d] * wr[d] + cr[d] * wr[D_ + d] + xr[d] * wr[2 * D_ + d];
    pgen[bt] = 1.0f / (1.0f + __expf(-acc));
}

// ---------------------------------------------------------------------------
// Per-row (over V) softmax statistics: rmax, rsum
// ---------------------------------------------------------------------------
__global__ __launch_bounds__(256)
void k_vstats(const float* __restrict__ logits,
              float* __restrict__ rmax, float* __restrict__ rsum) {
    int bt = blockIdx.x;
    const float* row = logits + (size_t)bt * V_;
    __shared__ float red[256];
    int tid = threadIdx.x;

    float m = -1e30f;
    for (int v = tid; v < V_; v += 256) m = fmaxf(m, row[v]);
    red[tid] = m; __syncthreads();
    for (int o = 128; o > 0; o >>= 1) {
        if (tid < o) red[tid] = fmaxf(red[tid], red[tid + o]);
        __syncthreads();
    }
    m = red[0]; __syncthreads();

    float sum = 0.0f;
    for (int v = tid; v < V_; v += 256) sum += __expf(row[v] - m);
    red[tid] = sum; __syncthreads();
    for (int o = 128; o > 0; o >>= 1) {
        if (tid < o) red[tid] += red[tid + o];
        __syncthreads();
    }
    if (tid == 0) { rmax[bt] = m; rsum[bt] = red[0]; }
}

// ---------------------------------------------------------------------------
// out0[b,t,v] = p_gen * softmax(logits), 128-bit vectorized, NT load + NT store
// (p_ctx term added afterwards by k_scatter via L2 atomics — still coherent)
// ---------------------------------------------------------------------------
__global__ void k_final(const v4f* __restrict__ logits4,
                        const float* __restrict__ rmax,
                        const float* __restrict__ rsum,
                        const float* __restrict__ pgen,
                        v4f* __restrict__ out4) {
    size_t i = (size_t)blockIdx.x * blockDim.x + threadIdx.x;  // over B*T*V/4
    if (i >= (size_t)B_ * T_ * V_ / 4) return;
    size_t bt  = (i * 4) / V_;                                  // V divisible by 4
    float  m   = rmax[bt];
    float  inv = pgen[bt] / rsum[bt];
    v4f lg = __builtin_nontemporal_load(&logits4[i]);
    v4f r;
    r.x = __expf(lg.x - m) * inv;
    r.y = __expf(lg.y - m) * inv;
    r.z = __expf(lg.z - m) * inv;
    r.w = __expf(lg.w - m) * inv;
    __builtin_nontemporal_store(r, &out4[i]);
}

// ---------------------------------------------------------------------------
// out0[b,t,story[b,s]] += (1 - p_gen) * probs[b,t,s]
// ---------------------------------------------------------------------------
__global__ void k_scatter(const float* __restrict__ probs,
                          const int*   __restrict__ story,
                          const float* __restrict__ pgen,
                          float* __restrict__ out0) {
    int i = blockIdx.x * blockDim.x + threadIdx.x;        // over B*T*S
    if (i >= B_ * T_ * S_) return;
    int s  = i % S_;
    int bt = i / S_;
    int b  = bt / T_;
    int v  = story[b * S_ + s];
    atomicAdd(&out0[(size_t)bt * V_ + v], (1.0f - pgen[bt]) * probs[i]);
}

// ---------------------------------------------------------------------------
extern "C" void kernel_launch(void* const* d_in, const int* in_sizes, int n_in,
                              void* d_out, int out_size, void* d_ws, size_t ws_size,
                              hipStream_t stream) {
    const float* enc_hidden = (const float*)d_in[0];   // (1,B,D) -> h0 (B,D)
    const float* enc_out    = (const float*)d_in[1];   // (B,S,D)
    const int*   story      = (const int*)  d_in[2];   // (B,S)
    const int*   trg        = (const int*)  d_in[3];   // (B,NS,T)
    const float* shared_emb = (const float*)d_in[4];   // (V,D)
    const float* slot_emb   = (const float*)d_in[5];   // (1,D)
    const float* w_ih       = (const float*)d_in[6];   // (3D,D)
    const float* w_hh       = (const float*)d_in[7];   // (3D,D)
    const float* b_ih       = (const float*)d_in[8];   // (3D,)
    const float* b_hh       = (const float*)d_in[9];   // (3D,)
    const float* w_ratio_w  = (const float*)d_in[10];  // (1,3D)
    const float* w_ratio_b  = (const float*)d_in[11];  // (1,)
    const float* w_gate_w   = (const float*)d_in[12];  // (NG,D)
    const float* w_gate_b   = (const float*)d_in[13];  // (NG,)

    float* out = (float*)d_out;

    // Workspace layout (floats)
    float* ws = (float*)d_ws;
    float* X      = ws;                                   //  51200
    float* gi     = X      + (size_t)B_ * T_ * D_;        // 153600
    float* gh     = gi     + (size_t)B_ * T_ * 3 * D_;    //  19200
    float* h      = gh     + (size_t)B_ * 3 * D_;         //  51200
    float* scores = h      + (size_t)B_ * T_ * D_;        //  65536
    float* probs  = scores + (size_t)B_ * T_ * S_;        //  65536
    float* ctx    = probs  + (size_t)B_ * T_ * S_;        //  51200
    float* logits = ctx    + (size_t)B_ * T_ * D_;        // 4096000
    float* pgen   = logits + (size_t)B_ * T_ * V_;        //    128
    float* rmax   = pgen   + (size_t)B_ * T_;             //    128
    float* rsum   = rmax   + (size_t)B_ * T_;             //    128

    const unsigned long long slice0    = (unsigned long long)B_ * T_ * V_;   // 4,096,000
    const unsigned long long total_out = (unsigned long long)out_size;
    const unsigned long long gate_off  = (unsigned long long)NS_ * B_ * T_ * V_;

    // 0) zero everything AFTER slice 0 (slice 0 fully overwritten by k_final);
    //    NT stores avoid flushing shared_emb/logits out of L2.
    k_zero<<<8192, 256, 0, stream>>>((v4f*)(out + slice0),
                                     (total_out - slice0) / 4ull);

    // 1) build X
    k_build_x<<<(B_ * T_ * D_ + 255) / 256, 256, 0, stream>>>(shared_emb, slot_emb, trg, X);

    dim3 wblk(32, 4);

    // 2) gi = X @ w_ih^T + b_ih   (M=128, N=1200, K=400)
    {
        dim3 grid((1200 / 16 + 3) / 4, (B_ * T_) / 16);
        k_wmma_gemm<<<grid, wblk, 0, stream>>>(X, w_ih, b_ih, gi, B_ * T_, 3 * D_, D_);
    }
    // 3) gh = h0 @ w_hh^T + b_hh  (M=16, N=1200, K=400)
    {
        dim3 grid((1200 / 16 + 3) / 4, 1);
        k_wmma_gemm<<<grid, wblk, 0, stream>>>(enc_hidden, w_hh, b_hh, gh, B_, 3 * D_, D_);
    }
    // 4) GRU elementwise -> h
    k_gru<<<(B_ * T_ * D_ + 255) / 256, 256, 0, stream>>>(gi, gh, enc_hidden, h);

    // 5) scores, softmax over S, context (raw scores)
    k_scores<<<(B_ * T_ * S_ + 255) / 256, 256, 0, stream>>>(h, enc_out, scores);
    k_softmax_s<<<B_ * T_, 256, 0, stream>>>(scores, probs);
    k_context<<<(B_ * T_ * D_ + 255) / 256, 256, 0, stream>>>(scores, enc_out, ctx);

    // 6) gate0 -> out[gate_off .. gate_off+B*NG)
    k_gate<<<1, 64, 0, stream>>>(ctx, w_gate_w, w_gate_b, out, gate_off);

    // 7) p_gen
    k_pgen<<<1, 128, 0, stream>>>(h, ctx, X, w_ratio_w, w_ratio_b, pgen);

    // 8) logits = h @ shared_emb^T  (M=128, N=32000, K=400) — the big WMMA GEMM
    {
        dim3 grid((V_ / 16 + 3) / 4, (B_ * T_) / 16);
        k_wmma_gemm<<<grid, wblk, 0, stream>>>(h, shared_emb, nullptr, logits, B_ * T_, V_, D_);
    }

    // 9) vocab softmax stats, final blend (vectorized, NT), scatter of copy dist
    k_vstats<<<B_ * T_, 256, 0, stream>>>(logits, rmax, rsum);
    k_final<<<(int)((slice0 / 4 + 255) / 256), 256, 0, stream>>>(
        (const v4f*)logits, rmax, rsum, pgen, (v4f*)out);
    k_scatter<<<(B_ * T_ * S_ + 255) / 256, 256, 0, stream>>>(probs, story, pgen, out);
}